// LSTM_128849018960
// MI455X (gfx1250) — compile-verified
//
#include <hip/hip_runtime.h>
#include <math.h>

// Problem constants
#define SEQ   256
#define BATCH 128
#define INP   512
#define HID   1024
#define G4H   4096       // 4*HID
#define NCLS  18
#define BH    (BATCH*HID)

typedef float v2f __attribute__((ext_vector_type(2)));
typedef float v8f __attribute__((ext_vector_type(8)));

__device__ __forceinline__ float sigmoid_(float x) {
  return 1.0f / (1.0f + __expf(-x));
}
__device__ __forceinline__ float tanh_(float x) {
  float xc = fminf(fmaxf(x, -15.0f), 15.0f);
  float e  = __expf(2.0f * xc);
  return (e - 1.0f) / (e + 1.0f);
}

// ---------------------------------------------------------------------------
// Prep kernels
// ---------------------------------------------------------------------------

// w: (4096, K) row-major  ->  wt: (K, 4096) row-major (coalesced writes)
__global__ void transpose_w(const float* __restrict__ w,
                            float* __restrict__ wt, int K) {
  int idx = blockIdx.x * blockDim.x + threadIdx.x;   // grid covers K*4096
  int n = idx & 4095;
  int k = idx >> 12;
  if (k < K) wt[idx] = w[(size_t)n * K + k];
}

__global__ void bias_sum(const float* __restrict__ a,
                         const float* __restrict__ b,
                         float* __restrict__ o) {
  int i = blockIdx.x * blockDim.x + threadIdx.x;
  if (i < G4H) o[i] = a[i] + b[i];
}

__global__ void zero_f32(float* __restrict__ p, int n) {
  int i = blockIdx.x * blockDim.x + threadIdx.x;
  if (i < n) p[i] = 0.0f;
}

// ---------------------------------------------------------------------------
// One LSTM gate-GEMM + cell-update tile (16 h-cols x all 128 batch rows).
//   A1: (128, K1) row stride ld1 (x_t or lower-layer h_t), A2: (128, HID) h_prev
//   Wt: (KTOT, 4096) = [W_ih ; W_hh]^T, gate-major columns (i,f,g,o)
//   256 threads = 8 waves; wave w owns batch rows 16w..16w+15.
// ---------------------------------------------------------------------------
__device__ __forceinline__ void lstm_tile(
    const float* __restrict__ A1, int ld1, int K1,
    const float* __restrict__ A2,
    const float* __restrict__ Wt,
    const float* __restrict__ bs,
    int KTOT, float* __restrict__ c_io, float* __restrict__ h_out,
    int hcol0, float (*As)[129], float (*Bs)[72]) {

  const int tid  = threadIdx.x;
  const int wave = tid >> 5;
  const int lane = tid & 31;
  const int ln   = lane & 15;
  const int hi   = lane >> 4;           // half-wave K offset select
  const int m0   = wave * 16;           // batch-row base for this wave

  // Accumulators: 4 gates, each a 16x16 f32 tile, bias-initialized.
  v8f acc[4];
#pragma unroll
  for (int g = 0; g < 4; ++g) {
    float bv = bs[g * HID + hcol0 + ln];
#pragma unroll
    for (int e = 0; e < 8; ++e) acc[g][e] = bv;
  }

  const int rowb = tid >> 4;            // 0..15 (A staging)
  const int kqa  = (tid & 15) * 4;      // 0..60 step 4
  const int kkb  = tid >> 2;            // 0..63 (B staging)
  const int qb   = (tid & 3) * 4;       // 0..12 step 4

  for (int k0 = 0; k0 < KTOT; k0 += 64) {
    const float* src;
    int ld;
    if (k0 < K1) { src = A1 + k0;        ld = ld1; }
    else         { src = A2 + (k0 - K1); ld = HID; }
    __syncthreads();
    // ---- stage A[0:128][k0:k0+64] transposed into LDS ----
#pragma unroll
    for (int p = 0; p < 8; ++p) {
      int row = p * 16 + rowb;
      float4 v = *reinterpret_cast<const float4*>(src + (size_t)row * ld + kqa);
      As[kqa + 0][row] = v.x;
      As[kqa + 1][row] = v.y;
      As[kqa + 2][row] = v.z;
      As[kqa + 3][row] = v.w;
    }
    // ---- stage B chunk (4 gates x 16 cols x 64 K) into LDS, shared by all waves
    {
      const float* wrow = Wt + (size_t)(k0 + kkb) * G4H + hcol0 + qb;
#pragma unroll
      for (int g = 0; g < 4; ++g) {
        float4 v = *reinterpret_cast<const float4*>(wrow + g * HID);
        *reinterpret_cast<float4*>(&Bs[kkb][g * 16 + qb]) = v;
      }
    }
    __syncthreads();

    // ---- 16 K-steps of V_WMMA_F32_16X16X4_F32 per gate ----
#pragma unroll 4
    for (int kk = 0; kk < 64; kk += 4) {
      v2f a;
      a.x = As[kk + 2 * hi + 0][m0 + ln];       // A[m=ln][k = 2*hi + v]
      a.y = As[kk + 2 * hi + 1][m0 + ln];
#pragma unroll
      for (int g = 0; g < 4; ++g) {
        v2f b;
        b.x = Bs[kk + 2 * hi + 0][g * 16 + ln]; // B[k=2*hi+v][n=ln]
        b.y = Bs[kk + 2 * hi + 1][g * 16 + ln];
        acc[g] = __builtin_amdgcn_wmma_f32_16x16x4_f32(
            false, a, false, b, (short)0, acc[g], false, false);
      }
    }
  }

  // ---- LSTM cell update; C/D layout: reg r -> row m0 + r + 8*hi, col ln ----
  const int col = hcol0 + ln;
#pragma unroll
  for (int r = 0; r < 8; ++r) {
    int brow = m0 + r + 8 * hi;
    size_t off = (size_t)brow * HID + col;
    float ig = sigmoid_(acc[0][r]);
    float fg = sigmoid_(acc[1][r]);
    float gg = tanh_(acc[2][r]);
    float og = sigmoid_(acc[3][r]);
    float cn = fg * c_io[off] + ig * gg;
    c_io[off]  = cn;
    h_out[off] = og * tanh_(cn);
  }
}

// ---------------------------------------------------------------------------
// Pipelined step kernel: blocks 0..63 run layer0 step t, blocks 64..127 run
// layer1 step t-1 (independent within one launch -> halves serial launches,
// doubles in-flight workgroups).
// ---------------------------------------------------------------------------
__global__ __launch_bounds__(256) void lstm_step2(
    const float* __restrict__ input,
    const float* __restrict__ Wcat0, const float* __restrict__ Wcat1,
    const float* __restrict__ bsum0, const float* __restrict__ bsum1,
    float* __restrict__ h0a, float* __restrict__ h0b, float* __restrict__ c0,
    float* __restrict__ h1a, float* __restrict__ h1b, float* __restrict__ c1,
    int t) {

  __shared__ float As[64][129];   // padded: conflict-free transpose writes
  __shared__ float Bs[64][72];    // padded: half-waves hit disjoint bank groups

  const int half  = blockIdx.x >> 6;
  const int hcol0 = (blockIdx.x & 63) * 16;

  if (half == 0) {
    if (t >= SEQ) return;
    lstm_tile(input + (size_t)t * BATCH * INP, INP, INP,
              (t & 1) ? h0b : h0a,                 // h0[t&1]
              Wcat0, bsum0, INP + HID, c0,
              (t & 1) ? h0a : h0b,                 // h0[(t+1)&1]
              hcol0, As, Bs);
  } else {
    int u = t - 1;
    if (u < 0) return;
    lstm_tile((u & 1) ? h0a : h0b, HID, HID,       // h0[(u+1)&1] = layer0 h_u
              (u & 1) ? h1b : h1a,                 // h1[u&1]
              Wcat1, bsum1, 2 * HID, c1,
              (u & 1) ? h1a : h1b,                 // h1[(u+1)&1]
              hcol0, As, Bs);
  }
}

// ---------------------------------------------------------------------------
// FC (1024 -> 18) + log_softmax, one block per batch row
// ---------------------------------------------------------------------------
__global__ __launch_bounds__(256) void fc_logsoftmax(
    const float* __restrict__ h,      // (128, 1024)
    const float* __restrict__ fcw,    // (18, 1024)
    const float* __restrict__ fcb,    // (18)
    float* __restrict__ out) {        // (128, 18)
  int b = blockIdx.x;
  __shared__ float hrow[HID];
  __shared__ float partial[NCLS * 8];
  for (int i = threadIdx.x; i < HID; i += 256) hrow[i] = h[(size_t)b * HID + i];
  __syncthreads();
  if (threadIdx.x < NCLS * 8) {
    int cl = threadIdx.x >> 3, sub = threadIdx.x & 7;
    float s = 0.0f;
    const float* wr = fcw + (size_t)cl * HID + sub * 128;
    const float* hr = hrow + sub * 128;
    for (int i = 0; i < 128; ++i) s += hr[i] * wr[i];
    partial[threadIdx.x] = s;
  }
  __syncthreads();
  if (threadIdx.x == 0) {
    float logits[NCLS], mx = -1e30f;
    for (int cl = 0; cl < NCLS; ++cl) {
      float s = fcb[cl];
      for (int j = 0; j < 8; ++j) s += partial[cl * 8 + j];
      logits[cl] = s;
      mx = fmaxf(mx, s);
    }
    float se = 0.0f;
    for (int cl = 0; cl < NCLS; ++cl) se += expf(logits[cl] - mx);
    float lse = mx + logf(se);
    for (int cl = 0; cl < NCLS; ++cl) out[b * NCLS + cl] = logits[cl] - lse;
  }
}

// ---------------------------------------------------------------------------
// Launch
// ---------------------------------------------------------------------------
extern "C" void kernel_launch(void* const* d_in, const int* in_sizes, int n_in,
                              void* d_out, int out_size, void* d_ws, size_t ws_size,
                              hipStream_t stream) {
  (void)in_sizes; (void)n_in; (void)out_size; (void)ws_size;
  const float* input = (const float*)d_in[0];
  const float* w_ih0 = (const float*)d_in[1];
  const float* w_hh0 = (const float*)d_in[2];
  const float* b_ih0 = (const float*)d_in[3];
  const float* b_hh0 = (const float*)d_in[4];
  const float* w_ih1 = (const float*)d_in[5];
  const float* w_hh1 = (const float*)d_in[6];
  const float* b_ih1 = (const float*)d_in[7];
  const float* b_hh1 = (const float*)d_in[8];
  const float* fcw   = (const float*)d_in[9];
  const float* fcb   = (const float*)d_in[10];
  float* out = (float*)d_out;

  // Workspace layout (floats): ~62 MB total (L2-resident weights)
  float* ws    = (float*)d_ws;
  float* Wcat0 = ws;                               // 1536*4096
  float* Wcat1 = Wcat0 + (size_t)1536 * G4H;       // 2048*4096
  float* bsum0 = Wcat1 + (size_t)2048 * G4H;       // 4096
  float* bsum1 = bsum0 + G4H;                      // 4096
  float* st    = bsum1 + G4H;                      // 6 * BH states
  float* h0a = st;
  float* h0b = st + BH;
  float* c0  = st + 2 * BH;
  float* h1a = st + 3 * BH;
  float* h1b = st + 4 * BH;
  float* c1  = st + 5 * BH;

  // --- prep: transpose weights into [W_ih ; W_hh]^T, pre-sum biases, zero state
  transpose_w<<<((size_t)512  * G4H) / 256, 256, 0, stream>>>(w_ih0, Wcat0,                      512);
  transpose_w<<<((size_t)1024 * G4H) / 256, 256, 0, stream>>>(w_hh0, Wcat0 + (size_t)512 * G4H,  1024);
  transpose_w<<<((size_t)1024 * G4H) / 256, 256, 0, stream>>>(w_ih1, Wcat1,                      1024);
  transpose_w<<<((size_t)1024 * G4H) / 256, 256, 0, stream>>>(w_hh1, Wcat1 + (size_t)1024 * G4H, 1024);
  bias_sum<<<G4H / 256, 256, 0, stream>>>(b_ih0, b_hh0, bsum0);
  bias_sum<<<G4H / 256, 256, 0, stream>>>(b_ih1, b_hh1, bsum1);
  zero_f32<<<(6 * BH) / 256, 256, 0, stream>>>(st, 6 * BH);

  // --- pipelined serial scan: launch k runs L0(t=k) and L1(t=k-1) ---
  for (int k = 0; k <= SEQ; ++k) {
    lstm_step2<<<128, 256, 0, stream>>>(input, Wcat0, Wcat1, bsum0, bsum1,
                                        h0a, h0b, c0, h1a, h1b, c1, k);
  }

  // final h1 lives in h1[SEQ & 1] == h1a
  fc_logsoftmax<<<BATCH, 256, 0, stream>>>(h1a, fcw, fcb, out);
}